// QuantumLiquidCell_88965952569850
// MI455X (gfx1250) — compile-verified
//
#include <hip/hip_runtime.h>
#include <math.h>

typedef float v2f __attribute__((ext_vector_type(2)));
typedef float v4f __attribute__((ext_vector_type(4)));
typedef float v8f __attribute__((ext_vector_type(8)));
typedef int   v4i __attribute__((vector_size(16)));

#define AS1 __attribute__((address_space(1)))
#define AS3 __attribute__((address_space(3)))

#define B_DIM 4096
#define I_DIM 1024
#define F_DIM 2048
#define Q_DIM 4

#define TAU_MIN 5.0f
#define TAU_MAX 50.0f
#define BOOST   3.2f
#define DECO_RATE 0.01f
#define DT_Q (0.1f / BOOST)

// output layout (floats): [0, B*F) new_hidden | [B*F, 2*B*F) d_amp | then d_coh, d_ent, d_dec, d_tau
#define OUT_SCALARS (2 * B_DIM * F_DIM)

// workspace layout (floats): q_state[F] | prob_mean[F] | qact[F] | rate[F]
#define WS_QSTATE 0
#define WS_PMEAN  F_DIM
#define WS_QACT   (2 * F_DIM)
#define WS_RATE   (3 * F_DIM)

__device__ __forceinline__ float sigmoidf_(float x) { return 1.0f / (1.0f + expf(-x)); }

// async global -> LDS copy of 16 bytes (GLOBAL_LOAD_ASYNC_TO_LDS_B128, ASYNCcnt)
__device__ __forceinline__ void async_copy_b128(const float* gsrc, float* ldst) {
    __builtin_amdgcn_global_load_async_to_lds_b128(
        (AS1 v4i*)(AS1 const void*)gsrc, (AS3 v4i*)ldst, 0, 0);
}
__device__ __forceinline__ void wait_async0() {
    asm volatile("s_wait_asynccnt 0" ::: "memory");
}

// ---------------------------------------------------------------- kernel 1
__global__ void init_scalars_kernel(float* out, const int* time_step) {
    if (threadIdx.x == 0) {
        float t = (float)time_step[0];
        out[OUT_SCALARS + 0] = expf(-DECO_RATE * t);  // d_coh
        out[OUT_SCALARS + 1] = 0.0f;                  // d_ent accumulator
        out[OUT_SCALARS + 2] = 0.0f;                  // d_dec accumulator
        out[OUT_SCALARS + 3] = 0.0f;                  // d_tau accumulator
    }
}

// ---------------------------------------------------------------- kernel 2
// per-feature quantum state: q_state[j], prob_mean[j]
__global__ void qstate_kernel(const float* __restrict__ amp_real,
                              const float* __restrict__ amp_imag,
                              const float* __restrict__ phase,
                              const int* __restrict__ time_step,
                              float* __restrict__ ws) {
    int j = blockIdx.x * blockDim.x + threadIdx.x;
    if (j >= F_DIM) return;
    float t = (float)time_step[0];
    float sup_r = 0.f, sup_i = 0.f, psum = 0.f;
#pragma unroll
    for (int q = 0; q < Q_DIM; ++q) {
        float ph = phase[j * Q_DIM + q] + t * 0.1f;
        float rc = amp_real[j * Q_DIM + q] * cosf(ph);
        float ic = amp_imag[j * Q_DIM + q] * sinf(ph);
        sup_r += rc;
        sup_i += ic;
        psum += rc * rc + ic * ic;
    }
    ws[WS_QSTATE + j] = sup_r * expf(-sup_i * sup_i);
    ws[WS_PMEAN + j]  = psum * (1.0f / Q_DIM);
}

// ---------------------------------------------------------------- kernel 3
// Collapsed tau/deco: q_state_b has identical rows, so [B,F]@[F,F] -> [F]@[F,F].
__global__ void taudec_kernel(const float* __restrict__ W_tau,
                              const float* __restrict__ b_tau,
                              const float* __restrict__ W_dec,
                              const float* __restrict__ b_dec,
                              const int* __restrict__ time_step,
                              float* __restrict__ ws,
                              float* __restrict__ out) {
    __shared__ float qs[F_DIM];
    __shared__ float red_d[256];
    __shared__ float red_t[256];
    int tid = threadIdx.x;
    for (int i = tid; i < F_DIM; i += blockDim.x) qs[i] = ws[WS_QSTATE + i];
    __syncthreads();

    int j = blockIdx.x * blockDim.x + tid;
    float st = 0.f, sd = 0.f;
    for (int k = 0; k < F_DIM; ++k) {
        float q = qs[k];
        st = fmaf(q, W_tau[k * F_DIM + j], st);
        sd = fmaf(q, W_dec[k * F_DIM + j], sd);
    }
    float tau_q = (TAU_MIN + (TAU_MAX - TAU_MIN) * sigmoidf_(st + b_tau[j])) / BOOST;
    float deco  = sigmoidf_(sd + b_dec[j]);
    float t = (float)time_step[0];
    float coherent = expf(-DECO_RATE * t);

    ws[WS_RATE + j] = DT_Q / tau_q;
    ws[WS_QACT + j] = qs[j] * deco * coherent;

    red_d[tid] = deco;
    red_t[tid] = tau_q;
    __syncthreads();
    for (int s = 128; s > 0; s >>= 1) {
        if (tid < s) { red_d[tid] += red_d[tid + s]; red_t[tid] += red_t[tid + s]; }
        __syncthreads();
    }
    if (tid == 0) {
        atomicAdd(&out[OUT_SCALARS + 2], red_d[0] * (1.0f / F_DIM));
        atomicAdd(&out[OUT_SCALARS + 3], red_t[0] * (1.0f / F_DIM));
    }
}

// ---------------------------------------------------------------- kernel 4
// Fused triple GEMM + liquid-cell epilogue, double-buffered async->LDS pipeline.
// Macro-tile 128(M=batch) x 64(N=feat), 256 threads = 8 waves (wave32),
// each wave owns a 32x32 tile = 2x2 WMMA 16x16 subtiles.
// Phase 1: K=2048, A=hidden, B={primary, ent_mat} (shared A tile).
// Phase 2: K=1024, A=inputs, B=W_in.
#define BM 128
#define BN 64
#define KT 16
#define AS_STRIDE 20
#define BS_STRIDE 68

__global__ __launch_bounds__(256) void
liquid_gemm_kernel(const float* __restrict__ inputs,
                   const float* __restrict__ hidden,
                   const float* __restrict__ W_in,
                   const float* __restrict__ b_in,
                   const float* __restrict__ primary,
                   const float* __restrict__ ent_mat,
                   const float* __restrict__ corr,
                   const float* __restrict__ ws,
                   float* __restrict__ out) {
    __shared__ float As[2][BM][AS_STRIDE];
    __shared__ float Bs1[2][KT][BS_STRIDE];
    __shared__ float Bs2[2][KT][BS_STRIDE];
    __shared__ float red[256];

    const int tid  = threadIdx.x;
    const int lane = tid & 31;
    const int wave = tid >> 5;
    const int wm   = wave >> 1;   // 0..3
    const int wn   = wave & 1;    // 0..1
    const int hi   = lane >> 4;   // 0 or 1 (lane half)
    const int l15  = lane & 15;

    const int bm = blockIdx.y * BM;
    const int bn = blockIdx.x * BN;

    // per-thread tile-load coordinates (float4 granules)
    const int rowA = tid >> 2;          // 0..63 (second granule: +64)
    const int c4A  = (tid & 3) * 4;     // 0,4,8,12
    const int rowB = tid >> 4;          // 0..15
    const int c4B  = (tid & 15) * 4;    // 0..60

    const v8f vzero = {0.f, 0.f, 0.f, 0.f, 0.f, 0.f, 0.f, 0.f};
    v8f acc1[2][2], acc2[2][2], accW[2][2];
#pragma unroll
    for (int im = 0; im < 2; ++im)
#pragma unroll
        for (int in = 0; in < 2; ++in) {
            acc1[im][in] = vzero; acc2[im][in] = vzero; accW[im][in] = vzero;
        }

    // ---------------- Phase 1: hidden @ {primary, ent_mat}, K = F_DIM
    {
        const int nsteps = F_DIM / KT;
        // prologue: prefetch step 0 into buffer 0
        async_copy_b128(&hidden[(size_t)(bm + rowA) * F_DIM + c4A],       &As[0][rowA][c4A]);
        async_copy_b128(&hidden[(size_t)(bm + rowA + 64) * F_DIM + c4A],  &As[0][rowA + 64][c4A]);
        async_copy_b128(&primary[(size_t)rowB * F_DIM + bn + c4B],        &Bs1[0][rowB][c4B]);
        async_copy_b128(&ent_mat[(size_t)rowB * F_DIM + bn + c4B],        &Bs2[0][rowB][c4B]);

        for (int s = 0; s < nsteps; ++s) {
            wait_async0();
            __syncthreads();
            if (s + 1 < nsteps) {
                const int k0 = (s + 1) * KT;
                const int nb = (s + 1) & 1;
                async_copy_b128(&hidden[(size_t)(bm + rowA) * F_DIM + k0 + c4A],      &As[nb][rowA][c4A]);
                async_copy_b128(&hidden[(size_t)(bm + rowA + 64) * F_DIM + k0 + c4A], &As[nb][rowA + 64][c4A]);
                async_copy_b128(&primary[(size_t)(k0 + rowB) * F_DIM + bn + c4B],     &Bs1[nb][rowB][c4B]);
                async_copy_b128(&ent_mat[(size_t)(k0 + rowB) * F_DIM + bn + c4B],     &Bs2[nb][rowB][c4B]);
            }
            const int buf = s & 1;
#pragma unroll
            for (int kk = 0; kk < KT; kk += 4) {
                v2f afrag[2];
#pragma unroll
                for (int im = 0; im < 2; ++im) {
                    int m = wm * 32 + im * 16 + l15;
                    afrag[im] = *(const v2f*)&As[buf][m][kk + 2 * hi];
                }
#pragma unroll
                for (int in = 0; in < 2; ++in) {
                    int n = wn * 32 + in * 16 + l15;
                    v2f b1, b2;
                    b1.x = Bs1[buf][kk + 2 * hi][n];
                    b1.y = Bs1[buf][kk + 2 * hi + 1][n];
                    b2.x = Bs2[buf][kk + 2 * hi][n];
                    b2.y = Bs2[buf][kk + 2 * hi + 1][n];
#pragma unroll
                    for (int im = 0; im < 2; ++im) {
                        acc1[im][in] = __builtin_amdgcn_wmma_f32_16x16x4_f32(
                            false, afrag[im], false, b1, (short)0, acc1[im][in], false, false);
                        acc2[im][in] = __builtin_amdgcn_wmma_f32_16x16x4_f32(
                            false, afrag[im], false, b2, (short)0, acc2[im][in], false, false);
                    }
                }
            }
        }
    }

    // ---------------- Phase 2: inputs @ W_in, K = I_DIM
    {
        const int nsteps = I_DIM / KT;
        // prologue: buffer 0 (safe: all waves are past reading buffer 0 of phase 1)
        async_copy_b128(&inputs[(size_t)(bm + rowA) * I_DIM + c4A],      &As[0][rowA][c4A]);
        async_copy_b128(&inputs[(size_t)(bm + rowA + 64) * I_DIM + c4A], &As[0][rowA + 64][c4A]);
        async_copy_b128(&W_in[(size_t)rowB * F_DIM + bn + c4B],          &Bs1[0][rowB][c4B]);

        for (int s = 0; s < nsteps; ++s) {
            wait_async0();
            __syncthreads();
            if (s + 1 < nsteps) {
                const int k0 = (s + 1) * KT;
                const int nb = (s + 1) & 1;
                async_copy_b128(&inputs[(size_t)(bm + rowA) * I_DIM + k0 + c4A],      &As[nb][rowA][c4A]);
                async_copy_b128(&inputs[(size_t)(bm + rowA + 64) * I_DIM + k0 + c4A], &As[nb][rowA + 64][c4A]);
                async_copy_b128(&W_in[(size_t)(k0 + rowB) * F_DIM + bn + c4B],        &Bs1[nb][rowB][c4B]);
            }
            const int buf = s & 1;
#pragma unroll
            for (int kk = 0; kk < KT; kk += 4) {
                v2f afrag[2];
#pragma unroll
                for (int im = 0; im < 2; ++im) {
                    int m = wm * 32 + im * 16 + l15;
                    afrag[im] = *(const v2f*)&As[buf][m][kk + 2 * hi];
                }
#pragma unroll
                for (int in = 0; in < 2; ++in) {
                    int n = wn * 32 + in * 16 + l15;
                    v2f b1;
                    b1.x = Bs1[buf][kk + 2 * hi][n];
                    b1.y = Bs1[buf][kk + 2 * hi + 1][n];
#pragma unroll
                    for (int im = 0; im < 2; ++im) {
                        accW[im][in] = __builtin_amdgcn_wmma_f32_16x16x4_f32(
                            false, afrag[im], false, b1, (short)0, accW[im][in], false, false);
                    }
                }
            }
        }
    }

    // ---------------- Epilogue: liquid-cell update + mean|ent_rec| reduction
    const float* qact = ws + WS_QACT;
    const float* rate = ws + WS_RATE;
    float local_abs = 0.f;

#pragma unroll
    for (int in = 0; in < 2; ++in) {
        int j  = bn + wn * 32 + in * 16 + l15;       // feature column
        float cj = corr[j];
        float bj = b_in[j];
        float qa = qact[j];
        float rt = rate[j];
#pragma unroll
        for (int im = 0; im < 2; ++im) {
            int rbase = bm + wm * 32 + im * 16 + 8 * hi;  // M = v + 8*hi within subtile
#pragma unroll
            for (int v = 0; v < 8; ++v) {
                int r = rbase + v;                   // global batch row
                float e    = acc1[im][in][v] + acc2[im][in][v] * cj;
                float cls  = tanhf(accW[im][in][v] + bj + e);
                float comb = cls + qa;
                float h    = hidden[(size_t)r * F_DIM + j];
                float nh   = h + rt * (comb - h);
                __builtin_nontemporal_store(nh, &out[(size_t)r * F_DIM + j]);
                local_abs += fabsf(e);
            }
        }
    }

    red[tid] = local_abs;
    __syncthreads();
    for (int s = 128; s > 0; s >>= 1) {
        if (tid < s) red[tid] += red[tid + s];
        __syncthreads();
    }
    if (tid == 0)
        atomicAdd(&out[OUT_SCALARS + 1], red[0] * (1.0f / ((float)B_DIM * (float)F_DIM)));
}

// ---------------------------------------------------------------- kernel 5
// d_amp[b, j] = prob_mean[j] broadcast over batch
__global__ void damp_fill_kernel(const float* __restrict__ ws, float* __restrict__ out) {
    const float* pm = ws + WS_PMEAN;
    int row = blockIdx.x;                 // one block per batch row
    int j4  = threadIdx.x * 4;            // 512 threads * 4 = 2048 = F_DIM
    v4f v;
    v.x = pm[j4 + 0]; v.y = pm[j4 + 1]; v.z = pm[j4 + 2]; v.w = pm[j4 + 3];
    __builtin_nontemporal_store(v, (v4f*)&out[(size_t)B_DIM * F_DIM + (size_t)row * F_DIM + j4]);
}

// ---------------------------------------------------------------- launch
extern "C" void kernel_launch(void* const* d_in, const int* in_sizes, int n_in,
                              void* d_out, int out_size, void* d_ws, size_t ws_size,
                              hipStream_t stream) {
    const float* inputs   = (const float*)d_in[0];
    const float* hidden   = (const float*)d_in[1];
    const float* W_in     = (const float*)d_in[2];
    const float* b_in     = (const float*)d_in[3];
    const float* amp_real = (const float*)d_in[4];
    const float* amp_imag = (const float*)d_in[5];
    const float* phase    = (const float*)d_in[6];
    const float* primary  = (const float*)d_in[7];
    const float* ent_mat  = (const float*)d_in[8];
    const float* corr     = (const float*)d_in[9];
    const float* W_tau    = (const float*)d_in[10];
    const float* b_tau    = (const float*)d_in[11];
    const float* W_dec    = (const float*)d_in[12];
    const float* b_dec    = (const float*)d_in[13];
    const int*   tstep    = (const int*)d_in[14];

    float* out = (float*)d_out;
    float* ws  = (float*)d_ws;

    init_scalars_kernel<<<1, 64, 0, stream>>>(out, tstep);
    qstate_kernel<<<F_DIM / 256, 256, 0, stream>>>(amp_real, amp_imag, phase, tstep, ws);
    taudec_kernel<<<F_DIM / 256, 256, 0, stream>>>(W_tau, b_tau, W_dec, b_dec, tstep, ws, out);

    dim3 grid(F_DIM / BN, B_DIM / BM);  // (32, 32)
    liquid_gemm_kernel<<<grid, 256, 0, stream>>>(inputs, hidden, W_in, b_in,
                                                 primary, ent_mat, corr, ws, out);

    damp_fill_kernel<<<B_DIM, F_DIM / 4, 0, stream>>>(ws, out);
}